// Net_40089224741482
// MI455X (gfx1250) — compile-verified
//
#include <hip/hip_runtime.h>
#include <hip/hip_bf16.h>
#include <math.h>

typedef __attribute__((ext_vector_type(2))) float v2f;
typedef __attribute__((ext_vector_type(8))) float v8f;
typedef int v4i_ __attribute__((vector_size(16)));   // matches builtin param type

#define BK 16

// ---------------------------------------------------------------------------
// CDNA5 async copy: global -> LDS, 16B per lane, tracked by ASYNCcnt.
// ---------------------------------------------------------------------------
__device__ __forceinline__ void async_b128(const float* g, float* l) {
#if __has_builtin(__builtin_amdgcn_global_load_async_to_lds_b128)
    __builtin_amdgcn_global_load_async_to_lds_b128(
        (__attribute__((address_space(1))) v4i_*)g,
        (__attribute__((address_space(3))) v4i_*)l, 0, 0);
#else
    unsigned laddr = (unsigned)(unsigned long long)
        (__attribute__((address_space(3))) float*)l;
    asm volatile("global_load_async_to_lds_b128 %0, %1, off"
                 :: "v"(laddr), "v"(g) : "memory");
#endif
}

__device__ __forceinline__ void wait_async0() {
#if __has_builtin(__builtin_amdgcn_s_wait_asynccnt)
    __builtin_amdgcn_s_wait_asynccnt(0);
#else
    asm volatile("s_wait_asynccnt 0x0" ::: "memory");
#endif
}

// ---------------------------------------------------------------------------
// fp32 WMMA GEMM: C[M,N] = act(A[M,K] @ W[K,N] + bias[N])
// 256 threads = 8 waves, block tile 128x64, wave tile 16x64.
// A: async global->LDS copies (double-buffered, padded rows).
// B: register-staged, k-pair interleaved so fragments are single ds_load_b64.
// One barrier per K-step.
// ---------------------------------------------------------------------------
__global__ __launch_bounds__(256) void gemm_wmma_f32(
    const float* __restrict__ A, const float* __restrict__ W,
    const float* __restrict__ bias, float* __restrict__ C,
    int M, int N, int K, int do_relu)
{
    __shared__ float As[2][128][BK + 4];   // padded: conflict-free b64 reads
    __shared__ float Bs[2][BK / 2][64][2]; // [buf][k-pair][n][k parity]

    const int tid  = threadIdx.x;
    const int wave = tid >> 5;
    const int lane = tid & 31;
    const int half = lane >> 4;
    const int mloc = lane & 15;
    const int bm = blockIdx.y * 128;
    const int bn = blockIdx.x * 64;

    const int ar = tid >> 2;               // A staging row (and +64)
    const int ac = (tid & 3) << 2;
    const int bkp = tid >> 5;              // B staging: wave w -> k-pair w
    const int bcc = lane << 1;             // 2 cols per thread

    v8f acc[4] = {};
    float2 pb0, pb1;

    auto asyncA = [&](int k0, int buf) {
        async_b128(A + (size_t)(bm + ar) * K + k0 + ac,      &As[buf][ar][ac]);
        async_b128(A + (size_t)(bm + 64 + ar) * K + k0 + ac, &As[buf][64 + ar][ac]);
    };
    auto gloadB = [&](int k0) {
        pb0 = *(const float2*)(W + (size_t)(k0 + 2 * bkp) * N + bn + bcc);
        pb1 = *(const float2*)(W + (size_t)(k0 + 2 * bkp + 1) * N + bn + bcc);
    };
    auto storeB = [&](int buf) {
        float4 v = {pb0.x, pb1.x, pb0.y, pb1.y};
        *(float4*)(&Bs[buf][bkp][bcc][0]) = v;
    };

    asyncA(0, 0);
    gloadB(0);
    storeB(0);
    wait_async0();
    __syncthreads();

    const int nIter = K / BK;
    for (int it = 0; it < nIter; ++it) {
        const int cur = it & 1;
        if (it + 1 < nIter) {              // prefetch next tile
            asyncA((it + 1) * BK, cur ^ 1);
            gloadB((it + 1) * BK);
        }
        #pragma unroll
        for (int ks = 0; ks < BK; ks += 4) {
            const int k2 = (ks >> 1) + half;
            v2f a = *(const v2f*)(&As[cur][(wave << 4) + mloc][ks + (half << 1)]);
            #pragma unroll
            for (int j = 0; j < 4; ++j) {
                v2f b = *(const v2f*)(&Bs[cur][k2][(j << 4) + mloc][0]);
                acc[j] = __builtin_amdgcn_wmma_f32_16x16x4_f32(
                    false, a, false, b, (short)0, acc[j], false, false);
            }
        }
        if (it + 1 < nIter) storeB(cur ^ 1);
        wait_async0();
        __syncthreads();
    }

    #pragma unroll
    for (int j = 0; j < 4; ++j) {
        const int n = bn + (j << 4) + mloc;
        const float bv = bias ? bias[n] : 0.0f;
        #pragma unroll
        for (int r = 0; r < 8; ++r) {
            const int m = bm + (wave << 4) + r + (half << 3);
            float v = acc[j][r] + bv;
            if (do_relu) v = fmaxf(v, 0.0f);
            C[(size_t)m * N + n] = v;
        }
    }
}

// ---------------------------------------------------------------------------
// Fused expert heads. Block = 128 batch rows x 4 experts (128 einsum cols).
// 256 threads = 8 waves, 8 accumulators each. Same async-A / interleaved-B
// double-buffered pipeline. Epilogue: tanh -> diff vs action -> ||.||^2
// (16-lane xor reduce) -> sqrt -> dist[b,n]. Centroids never hit memory.
// ---------------------------------------------------------------------------
__global__ __launch_bounds__(256) void expert_dist_wmma(
    const float* __restrict__ h, const float* __restrict__ Wexp,
    const float* __restrict__ bexp, const float* __restrict__ act,
    float* __restrict__ dist, int B_, int H_, int Nn, int A_)
{
    __shared__ float As[2][128][BK + 4];
    __shared__ float Bs[2][BK / 2][128][2];

    const int tid  = threadIdx.x;
    const int wave = tid >> 5;
    const int lane = tid & 31;
    const int half = lane >> 4;
    const int mloc = lane & 15;
    const int n0 = blockIdx.x * 4;
    const int bm = blockIdx.y * 128;

    const int ar = tid >> 2;
    const int ac = (tid & 3) << 2;
    const int bkp = tid >> 5;              // k-pair = wave
    const int bc  = lane << 2;             // 4 cols; expert = bc>>5, a = bc&31
    const float* Wn = Wexp + ((size_t)(n0 + (bc >> 5)) * H_) * A_ + (bc & 31);

    v8f acc[8] = {};
    float4 pb0, pb1;

    auto asyncA = [&](int k0, int buf) {
        async_b128(h + (size_t)(bm + ar) * H_ + k0 + ac,      &As[buf][ar][ac]);
        async_b128(h + (size_t)(bm + 64 + ar) * H_ + k0 + ac, &As[buf][64 + ar][ac]);
    };
    auto gloadB = [&](int k0) {
        pb0 = *(const float4*)(Wn + (size_t)(k0 + 2 * bkp) * A_);
        pb1 = *(const float4*)(Wn + (size_t)(k0 + 2 * bkp + 1) * A_);
    };
    auto storeB = [&](int buf) {
        float4 lo = {pb0.x, pb1.x, pb0.y, pb1.y};
        float4 hi = {pb0.z, pb1.z, pb0.w, pb1.w};
        *(float4*)(&Bs[buf][bkp][bc][0])     = lo;
        *(float4*)(&Bs[buf][bkp][bc + 2][0]) = hi;
    };

    asyncA(0, 0);
    gloadB(0);
    storeB(0);
    wait_async0();
    __syncthreads();

    const int nIter = H_ / BK;
    for (int it = 0; it < nIter; ++it) {
        const int cur = it & 1;
        if (it + 1 < nIter) {
            asyncA((it + 1) * BK, cur ^ 1);
            gloadB((it + 1) * BK);
        }
        #pragma unroll
        for (int ks = 0; ks < BK; ks += 4) {
            const int k2 = (ks >> 1) + half;
            v2f a = *(const v2f*)(&As[cur][(wave << 4) + mloc][ks + (half << 1)]);
            #pragma unroll
            for (int j = 0; j < 8; ++j) {
                v2f b = *(const v2f*)(&Bs[cur][k2][(j << 4) + mloc][0]);
                acc[j] = __builtin_amdgcn_wmma_f32_16x16x4_f32(
                    false, a, false, b, (short)0, acc[j], false, false);
            }
        }
        if (it + 1 < nIter) storeB(cur ^ 1);
        wait_async0();
        __syncthreads();
    }

    #pragma unroll
    for (int e = 0; e < 4; ++e) {
        const int n = n0 + e;
        const float be0 = bexp[(size_t)n * A_ + mloc];
        const float be1 = bexp[(size_t)n * A_ + 16 + mloc];
        #pragma unroll
        for (int r = 0; r < 8; ++r) {
            const int m = bm + (wave << 4) + r + (half << 3);
            const float d0 = tanhf(acc[2 * e][r] + be0)     - act[(size_t)m * A_ + mloc];
            const float d1 = tanhf(acc[2 * e + 1][r] + be1) - act[(size_t)m * A_ + 16 + mloc];
            float s = d0 * d0 + d1 * d1;
            s += __shfl_xor(s, 1);
            s += __shfl_xor(s, 2);
            s += __shfl_xor(s, 4);
            s += __shfl_xor(s, 8);
            if (mloc == 0) dist[(size_t)m * Nn + n] = sqrtf(s + 0.01f);
        }
    }
}

// ---------------------------------------------------------------------------
// out[b] = sum_n softmax(-dist[b,:])_n * values[b,n]   (N = 256, 1 block/row)
// ---------------------------------------------------------------------------
__global__ __launch_bounds__(256) void softmax_dot(
    const float* __restrict__ dist, const float* __restrict__ values,
    float* __restrict__ out, int Nn)
{
    const int b = blockIdx.x;
    const int t = threadIdx.x;
    const int wave = t >> 5, lane = t & 31;
    __shared__ float rmax[8], rse[8], rsn[8];

    const float logit = -dist[(size_t)b * Nn + t];
    const float val   = values[(size_t)b * Nn + t];

    float m = logit;
    #pragma unroll
    for (int o = 16; o > 0; o >>= 1) m = fmaxf(m, __shfl_xor(m, o));
    if (lane == 0) rmax[wave] = m;
    __syncthreads();
    m = rmax[0];
    #pragma unroll
    for (int i = 1; i < 8; ++i) m = fmaxf(m, rmax[i]);

    const float e = expf(logit - m);
    float se = e, sn = e * val;
    #pragma unroll
    for (int o = 16; o > 0; o >>= 1) {
        se += __shfl_xor(se, o);
        sn += __shfl_xor(sn, o);
    }
    if (lane == 0) { rse[wave] = se; rsn[wave] = sn; }
    __syncthreads();
    if (t == 0) {
        float SE = 0.0f, SN = 0.0f;
        #pragma unroll
        for (int i = 0; i < 8; ++i) { SE += rse[i]; SN += rsn[i]; }
        out[b] = SN / SE;
    }
}

// ---------------------------------------------------------------------------
extern "C" void kernel_launch(void* const* d_in, const int* in_sizes, int n_in,
                              void* d_out, int out_size, void* d_ws, size_t ws_size,
                              hipStream_t stream) {
    const int B = 4096, S = 128, A = 32, H = 1024, N = 256;

    const float* s    = (const float*)d_in[0];
    const float* a    = (const float*)d_in[1];
    const float* Wv1  = (const float*)d_in[2];
    const float* bv1  = (const float*)d_in[3];
    const float* Wv2  = (const float*)d_in[4];
    const float* bv2  = (const float*)d_in[5];
    const float* Wv3  = (const float*)d_in[6];
    const float* bv3  = (const float*)d_in[7];
    const float* Wv4  = (const float*)d_in[8];
    const float* bv4  = (const float*)d_in[9];
    const float* Wl1  = (const float*)d_in[10];
    const float* bl1  = (const float*)d_in[11];
    const float* Wexp = (const float*)d_in[12];
    const float* bexp = (const float*)d_in[13];
    float* out = (float*)d_out;

    // Workspace (floats): buf0[B*H], buf1[B*H], values[B*N], dist[B*N] = 40MB
    float* buf0   = (float*)d_ws;
    float* buf1   = buf0 + (size_t)B * H;
    float* values = buf1 + (size_t)B * H;
    float* dist   = values + (size_t)B * N;

    dim3 blk(256);
    // Value MLP: S->H->H->H->N
    gemm_wmma_f32<<<dim3(H / 64, B / 128), blk, 0, stream>>>(s,    Wv1, bv1, buf0,   B, H, S, 1);
    gemm_wmma_f32<<<dim3(H / 64, B / 128), blk, 0, stream>>>(buf0, Wv2, bv2, buf1,   B, H, H, 1);
    gemm_wmma_f32<<<dim3(H / 64, B / 128), blk, 0, stream>>>(buf1, Wv3, bv3, buf0,   B, H, H, 1);
    gemm_wmma_f32<<<dim3(N / 64, B / 128), blk, 0, stream>>>(buf0, Wv4, bv4, values, B, N, H, 0);
    // Location hidden: h = relu(s @ Wl1 + bl1)  (into buf1)
    gemm_wmma_f32<<<dim3(H / 64, B / 128), blk, 0, stream>>>(s,    Wl1, bl1, buf1,   B, H, S, 1);
    // Fused expert heads -> dist[B,N]
    expert_dist_wmma<<<dim3(N / 4, B / 128), blk, 0, stream>>>(buf1, Wexp, bexp, a, dist, B, H, N, A);
    // Softmax over N + weighted sum -> out[B,1]
    softmax_dot<<<dim3(B), dim3(256), 0, stream>>>(dist, values, out, N);
}